// YOLOV8LabelEncoder_76562087018509
// MI455X (gfx1250) — compile-verified
//
#include <hip/hip_runtime.h>
#include <cstdint>

// ---------------------------------------------------------------------------
// YOLOv8 label encoder (task-aligned assigner) for MI455X (gfx1250).
// Memory-bound (~180 MB -> ~7.7 us roofline @ 23.3 TB/s). No matmul structure
// worth WMMA (one-hot GEMM would ADD ~137 MB of HBM traffic); instead we use
// the CDNA5 Tensor Data Mover to double-buffer box tiles into LDS
// (tensor_load_to_lds + s_wait_tensorcnt) and global_prefetch for the strided
// score gather. wave32: TDM issued by wave 0 (threadIdx.x < 32) only.
// ---------------------------------------------------------------------------

#define Bc 32
#define Ac 8400
#define Cc 80
#define Mc 64
#define Kc 10
#define CH 512                 // anchors per TDM chunk (8 KB of box data)
#define NCH ((Ac + CH - 1) / CH)   // 17

typedef unsigned int u32;
typedef unsigned long long u64;

typedef __attribute__((ext_vector_type(4))) unsigned int v4u;
typedef __attribute__((ext_vector_type(8))) int v8i;
typedef __attribute__((ext_vector_type(4))) int v4i;

// ---- Tensor Data Mover: 1-D contiguous copy global -> LDS ------------------
// D# per CDNA5 ISA 8.3/8.4: group0 = {count=1, lds_addr, global_addr[56:0],
// type=2}; group1 = {wgmask=0, data_size=4B, tensor_dim0 (OOB zero-fill),
// tile_dim0}.  Groups 2/3 unused (<=2D tensor).
__device__ __forceinline__ void tdm_load(const void* g, u32 lds_off,
                                         u32 valid_elems, u32 tile_elems) {
  u64 ga = (u64)(uintptr_t)g;
  v4u g0;
  g0[0] = 1u;                                       // count=1, user mode
  g0[1] = lds_off;                                  // lds_addr
  g0[2] = (u32)ga;                                  // global_addr[31:0]
  g0[3] = (u32)((ga >> 32) & 0x01FFFFFFull) | (2u << 30);  // addr[56:32]|type=2
  v8i g1;
  g1[0] = (int)(2u << 16);                          // data_size = 4 bytes
  g1[1] = (int)((valid_elems & 0xFFFFu) << 16);     // tensor_dim0[15:0]
  g1[2] = (int)((valid_elems >> 16) & 0xFFFFu);     // tensor_dim0[31:16]
  g1[3] = (int)((tile_elems & 0xFFFFu) << 16);      // tile_dim0
  g1[4] = 0; g1[5] = 0; g1[6] = 0; g1[7] = 0;
  v4i z = (v4i)0;
#if defined(__clang_major__) && (__clang_major__ >= 23)
  v8i z8 = (v8i)0;
  __builtin_amdgcn_tensor_load_to_lds(g0, g1, z, z, z8, 0);
#else
  __builtin_amdgcn_tensor_load_to_lds(g0, g1, z, z, 0);
#endif
}

// ---- CIoU exactly as reference (KEPS on h terms, atan aspect penalty) ------
__device__ __forceinline__ float ciou_f(float gx0, float gy0, float gx1, float gy1,
                                        float gw, float gh, float gatan,
                                        float px0, float py0, float px1, float py1) {
  const float KEPS = 1e-7f;
  float w2 = px1 - px0;
  float h2 = py1 - py0 + KEPS;
  float iw = fmaxf(fminf(gx1, px1) - fmaxf(gx0, px0), 0.f);
  float ih = fmaxf(fminf(gy1, py1) - fmaxf(gy0, py0), 0.f);
  float inter = iw * ih;
  float uni = gw * gh + w2 * h2 - inter + KEPS;
  float iou = inter / uni;
  float cw = fmaxf(gx1, px1) - fminf(gx0, px0);
  float ch = fmaxf(gy1, py1) - fminf(gy0, py0);
  float c2 = cw * cw + ch * ch + KEPS;
  float dx = px0 + px1 - gx0 - gx1;
  float dy = py0 + py1 - gy0 - gy1;
  float rho2 = (dx * dx + dy * dy) * 0.25f;
  float da = atanf(w2 / h2) - gatan;
  float v = 0.40528473456f * da * da;               // 4/pi^2
  float alpha = v / (v - iou + (1.f + KEPS));
  return iou - (rho2 / c2 + v * alpha);
}

// monotone float->uint mapping (total order incl. negatives)
__device__ __forceinline__ u32 forder(float f) {
  u32 u = __float_as_uint(f);
  return (u & 0x80000000u) ? ~u : (u | 0x80000000u);
}

// ---- workspace zero-init ---------------------------------------------------
__global__ void initws(uint4* __restrict__ ws, int n16) {
  int i = blockIdx.x * blockDim.x + threadIdx.x;
  if (i < n16) ws[i] = make_uint4(0u, 0u, 0u, 0u);
}

// ---- Stage 1: per (b,m) alignment row + top-K + sparse scatter -------------
__global__ __launch_bounds__(256) void stage1(
    const float* __restrict__ scores, const float* __restrict__ dbox,
    const float* __restrict__ anc, const int* __restrict__ glabel,
    const float* __restrict__ gbox, const unsigned char* __restrict__ gmask,
    u64* __restrict__ bestkey, u64* __restrict__ mmask, u32* __restrict__ normv) {
  const int m = blockIdx.x, b = blockIdx.y;
  const int tid = threadIdx.x;
  if (!gmask[b * Mc + m]) return;   // masked GT: whole alignment row is zero

  __shared__ float align_s[Ac];            // 33.6 KB full alignment row
  __shared__ float boxbuf[2][CH * 4];      // 16 KB double-buffered TDM target
  __shared__ float red_v[256];
  __shared__ int   red_i[256];
  __shared__ float cand_m[Kc];
  __shared__ int   cand_i[Kc];
  __shared__ float cand_o[Kc];
  __shared__ int s_cnt, s_stop;
  __shared__ float s_maxov;

  const size_t gm = (size_t)b * Mc + m;
  const float gx0 = gbox[gm * 4 + 0], gy0 = gbox[gm * 4 + 1];
  const float gx1 = gbox[gm * 4 + 2], gy1 = gbox[gm * 4 + 3];
  const float gw = gx1 - gx0;
  const float gh = gy1 - gy0 + 1e-7f;
  const float gatan = atanf(gw / gh);
  int lbl = glabel[gm]; if (lbl < 0) lbl = 0;
  const float* srow = scores + (size_t)b * Ac * Cc + lbl;
  const float* gsrc = dbox + (size_t)b * Ac * 4;

  const bool wave0 = (tid < 32);           // wave32: first wave issues TDM
  const u32 lds_base = (u32)(uintptr_t)&boxbuf[0][0];

  if (wave0) tdm_load(gsrc, lds_base, (u32)(Ac * 4), CH * 4);
  if (tid == 0) { s_cnt = 0; s_stop = 0; }

  for (int c = 0; c < NCH; ++c) {
    const int a0 = c * CH;
    const int sel = c & 1;
    if (wave0) {
      if (c + 1 < NCH) {
        tdm_load(gsrc + (size_t)(a0 + CH) * 4,
                 lds_base + (u32)(((c + 1) & 1) * CH * 4 * sizeof(float)),
                 (u32)((Ac - (a0 + CH)) * 4), CH * 4);
        __builtin_amdgcn_s_wait_tensorcnt(1);   // chunk c landed, c+1 in flight
      } else {
        __builtin_amdgcn_s_wait_tensorcnt(0);
      }
    }
    __syncthreads();
#pragma unroll
    for (int i = 0; i < CH / 256; ++i) {
      const int la = i * 256 + tid;
      const int a = a0 + la;
      if (a < Ac) {
        const float4 pb = ((const float4*)&boxbuf[sel][0])[la];   // ds_load_b128
        const float ax = anc[a * 2 + 0], ay = anc[a * 2 + 1];
        float val = 0.f;
        if (gx0 < ax && gy0 < ay && gx1 > ax && gy1 > ay) {
          const float ov = ciou_f(gx0, gy0, gx1, gy1, gw, gh, gatan,
                                  pb.x, pb.y, pb.z, pb.w);
          const float sc = srow[(size_t)a * Cc];
          const float o2 = ov * ov;
          val = sqrtf(sc) * (o2 * o2 * o2);      // score^0.5 * ciou^6
        }
        align_s[a] = val;
        if (a + CH < Ac)                          // hide the 320B-stride gather
          __builtin_prefetch(srow + (size_t)(a + CH) * Cc, 0, 0);
      }
    }
    __syncthreads();                              // before buf[sel] reuse
  }

  // top-K with first-index tie-break == jax.lax.top_k order
  for (int k = 0; k < Kc; ++k) {
    float bv = -2.f; int bi = 0;
    for (int a = tid; a < Ac; a += 256) {
      const float v = align_s[a];
      if (v > bv) { bv = v; bi = a; }
    }
    red_v[tid] = bv; red_i[tid] = bi;
    __syncthreads();
    for (int s = 128; s > 0; s >>= 1) {
      if (tid < s) {
        const float ov_ = red_v[tid + s];
        if (ov_ > red_v[tid] ||
            (ov_ == red_v[tid] && red_i[tid + s] < red_i[tid])) {
          red_v[tid] = ov_; red_i[tid] = red_i[tid + s];
        }
      }
      __syncthreads();
    }
    if (tid == 0) {
      if (red_v[0] > 0.f) {                        // cand_metric > 0 filter
        cand_m[k] = red_v[0]; cand_i[k] = red_i[0];
        align_s[red_i[0]] = -1.f;
        s_cnt = k + 1;
      } else s_stop = 1;
    }
    __syncthreads();
    if (s_stop) break;
  }

  const int cnt = s_cnt;
  if (cnt == 0) return;

  if (tid < cnt) {   // recompute CIoU for the <=10 winners (cheap)
    const int a = cand_i[tid];
    const float4 pb = *((const float4*)(dbox + ((size_t)b * Ac + a) * 4));
    cand_o[tid] = ciou_f(gx0, gy0, gx1, gy1, gw, gh, gatan,
                         pb.x, pb.y, pb.z, pb.w);
  }
  __syncthreads();
  if (tid == 0) {
    float mo = 0.f;                                // row max includes zeros
    for (int i = 0; i < cnt; ++i) mo = fmaxf(mo, cand_o[i]);
    s_maxov = mo;
  }
  __syncthreads();
  if (tid < cnt) {
    const int a = cand_i[tid];
    const size_t p = (size_t)b * Ac + a;
    // key: max overlap wins, ties -> smallest m (argmax over axis=1 semantics)
    const u64 key = ((u64)forder(cand_o[tid]) << 32) | (u32)(~(u32)m);
    atomicMax(&bestkey[p], key);
    atomicOr(&mmask[p], 1ull << m);
    const float nv = cand_m[tid] * s_maxov / (cand_m[0] + 1e-9f);
    atomicMax(&normv[p], __float_as_uint(nv));     // nv >= 0: uint order == float order
  }
}

// ---- Stage 2: per (b,a) decode + stream outputs ----------------------------
__global__ __launch_bounds__(256) void stage2(
    const u64* __restrict__ bestkey, const u64* __restrict__ mmask,
    const u32* __restrict__ normv, const float* __restrict__ gbox,
    const int* __restrict__ glabel, float* __restrict__ out) {
  const int b = blockIdx.y;
  const int a = blockIdx.x * 256 + threadIdx.x;
  __shared__ float4 sbox[Mc];
  __shared__ int slbl[Mc];
  if (threadIdx.x < Mc) {
    sbox[threadIdx.x] = ((const float4*)gbox)[b * Mc + threadIdx.x];
    slbl[threadIdx.x] = glabel[b * Mc + threadIdx.x];
  }
  __syncthreads();
  if (a >= Ac) return;

  const size_t p = (size_t)b * Ac + a;
  const u64 key = bestkey[p];
  const u32 hi = (u32)(key >> 32);
  const bool pos = hi > 0x80000000u;               // max matched overlap > 0
  const int mdec = (int)(~(u32)key) & 63;
  int mm;
  if (pos) mm = mdec;
  else {
    const u64 inv = ~mmask[p];
    // argmax of a row of {matched ov<=0, zeros}: first zero -> smallest
    // unmatched m; all-matched edge case falls back to the packed argmax.
    mm = inv ? (int)__builtin_ctzll(inv) : mdec;
  }
  const float nv = __uint_as_float(normv[p]);

  float4 bl; int lbl;
  if (pos) { bl = sbox[mm]; lbl = slbl[mm]; }
  else { bl = make_float4(-1.f, -1.f, -1.f, -1.f); lbl = -1; }

  ((float4*)out)[p] = bl;                          // bbox_labels [B,A,4]

  float* crow = out + (size_t)Bc * Ac * 4 + p * (size_t)Cc;
#pragma unroll
  for (int g = 0; g < Cc / 4; ++g) {               // one-hot * norm_align
    float4 v;
    v.x = (4 * g + 0 == lbl) ? nv : 0.f;
    v.y = (4 * g + 1 == lbl) ? nv : 0.f;
    v.z = (4 * g + 2 == lbl) ? nv : 0.f;
    v.w = (4 * g + 3 == lbl) ? nv : 0.f;
    ((float4*)crow)[g] = v;
  }
  // reference quirk: (gt_match > 0), independent of the match mask
  out[(size_t)Bc * Ac * 4 + (size_t)Bc * Ac * Cc + p] = (mm > 0) ? 1.f : 0.f;
}

// ---------------------------------------------------------------------------
extern "C" void kernel_launch(void* const* d_in, const int* in_sizes, int n_in,
                              void* d_out, int out_size, void* d_ws, size_t ws_size,
                              hipStream_t stream) {
  (void)in_sizes; (void)n_in; (void)out_size; (void)ws_size;
  const float* scores  = (const float*)d_in[0];   // [B,A,C] f32
  const float* dbox    = (const float*)d_in[1];   // [B,A,4] f32
  const float* anchors = (const float*)d_in[2];   // [A,2]   f32
  const int*   glabel  = (const int*)d_in[3];     // [B,M]   i32
  const float* gbox    = (const float*)d_in[4];   // [B,M,4] f32
  const unsigned char* gmask = (const unsigned char*)d_in[5];  // [B,M,1] bool (1B)
  float* out = (float*)d_out;

  const size_t BA = (size_t)Bc * Ac;
  u64* bestkey = (u64*)d_ws;                       // [B*A] packed (ov, ~m)
  u64* mmask   = bestkey + BA;                     // [B*A] matched-m bitmask
  u32* normv   = (u32*)(mmask + BA);               // [B*A] norm-align bits
  const int n16 = (int)(BA * 20 / 16);             // 5.376 MB / 16B

  initws<<<(n16 + 255) / 256, 256, 0, stream>>>((uint4*)d_ws, n16);
  stage1<<<dim3(Mc, Bc), 256, 0, stream>>>(scores, dbox, anchors, glabel, gbox,
                                           gmask, bestkey, mmask, normv);
  stage2<<<dim3((Ac + 255) / 256, Bc), 256, 0, stream>>>(bestkey, mmask, normv,
                                                         gbox, glabel, out);
}